// VoronoiGCN_84988812853414
// MI455X (gfx1250) — compile-verified
//
#include <hip/hip_runtime.h>

#define NN 100000
#define NE 1600000
#define FIN 5
#define H 64
#define HC 32
#define NL 3
#define BN_EPS 1e-5f
#define BPAD 4            // LDS row padding: stride H+4=68 floats, keeps 8B align + no bank conflicts

typedef __attribute__((ext_vector_type(2))) float v2f;
typedef __attribute__((ext_vector_type(8))) float v8f;

// ---------------------------------------------------------------------------
// Degree / normalization coefficient kernels
// ---------------------------------------------------------------------------
__global__ void deg_init_kernel(float* __restrict__ deg) {
    int i = blockIdx.x * blockDim.x + threadIdx.x;
    if (i < NN) deg[i] = 1.0f;   // self-loop weight included
}

__global__ void deg_accum_kernel(const int* __restrict__ dst, float* __restrict__ deg) {
    int e = blockIdx.x * blockDim.x + threadIdx.x;
    if (e < NE) atomicAdd(&deg[dst[e]], 1.0f);
}

__global__ void dinv_kernel(float* __restrict__ deg_dinv) {
    int i = blockIdx.x * blockDim.x + threadIdx.x;
    if (i < NN) deg_dinv[i] = rsqrtf(deg_dinv[i]);   // in place: deg -> deg^{-1/2}
}

__global__ void edge_coef_kernel(const int* __restrict__ src, const int* __restrict__ dst,
                                 const float* __restrict__ dinv, float* __restrict__ coef) {
    int e = blockIdx.x * blockDim.x + threadIdx.x;
    if (e < NE) coef[e] = dinv[src[e]] * dinv[dst[e]];
}

// ---------------------------------------------------------------------------
// Input projection: h = relu(x @ W_in + b_in)   (N x 5) @ (5 x 64)
// ---------------------------------------------------------------------------
__global__ void input_proj_kernel(const float* __restrict__ x, const float* __restrict__ W,
                                  const float* __restrict__ b, float* __restrict__ h) {
    int idx = blockIdx.x * blockDim.x + threadIdx.x;
    if (idx >= NN * H) return;
    int n = idx >> 6;
    int c = idx & 63;
    float s = b[c];
#pragma unroll
    for (int k = 0; k < FIN; ++k) s += x[n * FIN + k] * W[k * H + c];
    h[idx] = fmaxf(s, 0.0f);
}

// ---------------------------------------------------------------------------
// WMMA fp32 GEMM, K fixed = 64: C(N x ncols) = A(N x 64) @ B(64 x ncols)
//   mode 0 (conv layer):  C = hw,  C2 = hw * dinv[n]^2 + bias   (fused agg init)
//   mode 1 (classifier):  C = relu(hw + bias)
// B is staged once per block into LDS *transposed* (sBt[col][k], padded row
// stride 68 floats -> 8B-aligned ds_load_b64, conflict-free).
// A fragment (16 x v2f) hoisted into registers as contiguous b64 loads, then a
// fully-unrolled chain of 16 V_WMMA_F32_16X16X4_F32.
// fp32 A-frag layout (ISA 7.12.2): lanes 0-15 hold M=0..15 with K={k,k+1},
// lanes 16-31 hold M=0..15 with K={k+2,k+3}. B/C/D rows striped across lanes.
// ---------------------------------------------------------------------------
__global__ void gemm_wmma_kernel(const float* __restrict__ A, const float* __restrict__ Bm,
                                 const float* __restrict__ bias, const float* __restrict__ dinv,
                                 float* __restrict__ C, float* __restrict__ C2,
                                 int nRowTiles, int ncols, int mode) {
    __shared__ float sBt[H][H + BPAD];          // max 64 cols x (64+4) floats = 17.4 KB

    // stage B transposed: global reads coalesced (idx-major), LDS writes 2-way max
    for (int idx = threadIdx.x; idx < H * ncols; idx += blockDim.x) {
        int k = idx / ncols;
        int c = idx - k * ncols;
        sBt[c][k] = Bm[idx];
    }
    __syncthreads();

    int wave   = blockIdx.x * (blockDim.x >> 5) + (threadIdx.x >> 5);
    int tilesX = ncols >> 4;
    int tr     = wave / tilesX;
    int tc     = wave - tr * tilesX;

    if (tr < nRowTiles) {                       // wave-uniform: EXEC all-1 at WMMA
        int lane = threadIdx.x & 31;
        int hi   = lane >> 4;                   // 0: K pair {0,1}; 1: K pair {2,3}
        int l    = lane & 15;
        int koff = hi ? 2 : 0;

        const float* Arow = A + (size_t)(tr * 16 + l) * H;
        int bcol = tc * 16 + l;

        // hoist A fragment: 16 contiguous 8-byte loads
        v2f af[16];
#pragma unroll
        for (int s = 0; s < 16; ++s) {
            int ka = 4 * s + koff;
            af[s].x = Arow[ka];
            af[s].y = Arow[ka + 1];
        }

        v8f acc = {};
#pragma unroll
        for (int s = 0; s < 16; ++s) {
            int ka = 4 * s + koff;
            v2f b;
            b.x = sBt[bcol][ka];
            b.y = sBt[bcol][ka + 1];
            acc = __builtin_amdgcn_wmma_f32_16x16x4_f32(
                false, af[s], false, b, (short)0, acc, false, false);
        }

        float bv = bias ? bias[bcol] : 0.0f;
#pragma unroll
        for (int j = 0; j < 8; ++j) {
            int m   = tr * 16 + j + (hi ? 8 : 0);   // VGPR j: rows j / j+8
            float v = acc[j];
            size_t o = (size_t)m * ncols + bcol;
            if (mode == 0) {
                C[o] = v;                           // hw (scatter source)
                float dd = dinv[m];
                C2[o] = v * (dd * dd) + bv;         // agg = self-loop + bias
            } else {
                C[o] = fmaxf(v + bv, 0.0f);         // classifier hidden
            }
        }
    }
}

// ---------------------------------------------------------------------------
// Scatter: agg[dst] += hw[src] * edge_coef.  One wave per edge: indices/coef
// are wave-uniform loads (one cacheline txn), gathers are coalesced float2,
// 2 global_atomic_add_f32 per lane (64 channels / 32 lanes).
// ---------------------------------------------------------------------------
__global__ void scatter_kernel(const float* __restrict__ hw, const float* __restrict__ coef,
                               const int* __restrict__ src, const int* __restrict__ dst,
                               float* __restrict__ agg) {
    int e = blockIdx.x * (blockDim.x >> 5) + (threadIdx.x >> 5);
    if (e >= NE) return;
    int lane = threadIdx.x & 31;
    int s  = src[e];
    int d  = dst[e];
    float cf = coef[e];
    const float2* hs = (const float2*)(hw + (size_t)s * H);
    float2 v = hs[lane];
    float* ad = agg + (size_t)d * H + lane * 2;
    atomicAdd(ad,     v.x * cf);
    atomicAdd(ad + 1, v.y * cf);
}

// ---------------------------------------------------------------------------
// BatchNorm: per-channel sum / sumsq partial reduction -> global atomics
// block = 256 threads = 64 channels x 4 row-groups, 128 rows per block
// ---------------------------------------------------------------------------
__global__ void zero_kernel(float* __restrict__ p, int n) {
    int i = blockIdx.x * blockDim.x + threadIdx.x;
    if (i < n) p[i] = 0.0f;
}

__global__ void bn_reduce_kernel(const float* __restrict__ agg,
                                 float* __restrict__ sum, float* __restrict__ sumsq) {
    __shared__ float s_sum[4][64];
    __shared__ float s_sq[4][64];
    int c = threadIdx.x & 63;
    int g = threadIdx.x >> 6;
    int base = blockIdx.x * 128;
    float ls = 0.0f, lq = 0.0f;
    for (int r = base + g; r < base + 128 && r < NN; r += 4) {
        float v = agg[(size_t)r * H + c];
        ls += v;
        lq += v * v;
    }
    s_sum[g][c] = ls;
    s_sq[g][c]  = lq;
    __syncthreads();
    if (g == 0) {
        ls = s_sum[0][c] + s_sum[1][c] + s_sum[2][c] + s_sum[3][c];
        lq = s_sq[0][c] + s_sq[1][c] + s_sq[2][c] + s_sq[3][c];
        atomicAdd(&sum[c], ls);
        atomicAdd(&sumsq[c], lq);
    }
}

__global__ void bn_finalize_kernel(const float* __restrict__ sum, const float* __restrict__ sumsq,
                                   const float* __restrict__ gamma, const float* __restrict__ beta,
                                   float* __restrict__ scale, float* __restrict__ shift) {
    int c = threadIdx.x;
    if (c >= H) return;
    float inv_n = 1.0f / (float)NN;
    float mean  = sum[c] * inv_n;
    float var   = sumsq[c] * inv_n - mean * mean;
    float rstd  = rsqrtf(var + BN_EPS);
    float sc    = gamma[c] * rstd;
    scale[c] = sc;
    shift[c] = beta[c] - mean * sc;
}

__global__ void bn_norm_relu_kernel(const float* __restrict__ agg,
                                    const float* __restrict__ scale, const float* __restrict__ shift,
                                    float* __restrict__ h) {
    int idx = blockIdx.x * blockDim.x + threadIdx.x;
    if (idx >= NN * H) return;
    int c = idx & 63;
    h[idx] = fmaxf(agg[idx] * scale[c] + shift[c], 0.0f);
}

// ---------------------------------------------------------------------------
// Final classifier: out[n] = z[n,:] @ W2 + b2   (z is N x 32, relu already done)
// ---------------------------------------------------------------------------
__global__ void classifier_out_kernel(const float* __restrict__ z, const float* __restrict__ W2,
                                      const float* __restrict__ b2, float* __restrict__ out) {
    int n = blockIdx.x * blockDim.x + threadIdx.x;
    if (n >= NN) return;
    float s = b2[0];
#pragma unroll
    for (int j = 0; j < HC; ++j) s += z[(size_t)n * HC + j] * W2[j];
    out[n] = s;
}

// ---------------------------------------------------------------------------
extern "C" void kernel_launch(void* const* d_in, const int* in_sizes, int n_in,
                              void* d_out, int out_size, void* d_ws, size_t ws_size,
                              hipStream_t stream) {
    (void)in_sizes; (void)n_in; (void)out_size; (void)ws_size;

    const float* x     = (const float*)d_in[0];
    const int*   ei    = (const int*)  d_in[1];
    const float* W_in  = (const float*)d_in[2];
    const float* b_in  = (const float*)d_in[3];
    const float* convW = (const float*)d_in[4];
    const float* convB = (const float*)d_in[5];
    const float* gamma = (const float*)d_in[6];
    const float* beta  = (const float*)d_in[7];
    const float* W1    = (const float*)d_in[8];
    const float* b1    = (const float*)d_in[9];
    const float* W2    = (const float*)d_in[10];
    const float* b2    = (const float*)d_in[11];
    float* out = (float*)d_out;

    const int* src = ei;          // edge_index row 0
    const int* dst = ei + NE;     // edge_index row 1

    // workspace layout (floats)
    float* ws = (float*)d_ws;
    size_t off = 0;
    float* h    = ws + off; off += (size_t)NN * H;
    float* hw   = ws + off; off += (size_t)NN * H;
    float* agg  = ws + off; off += (size_t)NN * H;
    float* z    = ws + off; off += (size_t)NN * HC;
    float* dinv = ws + off; off += NN;          // deg, then deg^{-1/2} in place
    float* coef = ws + off; off += NE;
    float* bnstat  = ws + off; off += 2 * H;    // [sum(64) | sumsq(64)]
    float* bnscale = ws + off; off += H;
    float* bnshift = ws + off; off += H;

    const int B = 256;
    int gN   = (NN + B - 1) / B;
    int gE   = (NE + B - 1) / B;
    int gNH  = (NN * H + B - 1) / B;
    int gBN  = (NN + 127) / 128;
    int gSc  = (NE + 7) / 8;                    // one wave (32 lanes) per edge, 8 waves/block

    // --- GCN normalization coefficients ---
    deg_init_kernel<<<gN, B, 0, stream>>>(dinv);
    deg_accum_kernel<<<gE, B, 0, stream>>>(dst, dinv);
    dinv_kernel<<<gN, B, 0, stream>>>(dinv);
    edge_coef_kernel<<<gE, B, 0, stream>>>(src, dst, dinv, coef);

    // --- input projection ---
    input_proj_kernel<<<gNH, B, 0, stream>>>(x, W_in, b_in, h);

    // --- GCN layers ---
    const int rowTiles = NN / 16;                       // 6250, exact
    int gGemm64 = (rowTiles * (H / 16) + 7) / 8;        // 8 waves / block, exact
    for (int i = 0; i < NL; ++i) {
        gemm_wmma_kernel<<<gGemm64, B, 0, stream>>>(h, convW + (size_t)i * H * H,
                                                    convB + (size_t)i * H, dinv,
                                                    hw, agg, rowTiles, H, 0);
        scatter_kernel<<<gSc, B, 0, stream>>>(hw, coef, src, dst, agg);

        zero_kernel<<<1, 2 * H, 0, stream>>>(bnstat, 2 * H);
        bn_reduce_kernel<<<gBN, B, 0, stream>>>(agg, bnstat, bnstat + H);
        bn_finalize_kernel<<<1, H, 0, stream>>>(bnstat, bnstat + H,
                                                gamma + (size_t)i * H, beta + (size_t)i * H,
                                                bnscale, bnshift);
        bn_norm_relu_kernel<<<gNH, B, 0, stream>>>(agg, bnscale, bnshift, h);
    }

    // --- classifier ---
    int gGemm32 = (rowTiles * (HC / 16) + 7) / 8;
    gemm_wmma_kernel<<<gGemm32, B, 0, stream>>>(h, W1, b1, nullptr, z, nullptr,
                                                rowTiles, HC, 1);
    classifier_out_kernel<<<gN, B, 0, stream>>>(z, W2, b2, out);
}